// EuclideanIAHMLoss_5746666242188
// MI455X (gfx1250) — compile-verified
//
#include <hip/hip_runtime.h>
#include <hip/hip_bf16.h>

typedef __attribute__((ext_vector_type(2))) float v2f;
typedef __attribute__((ext_vector_type(8))) float v8f;

#define N_CLASSES 40
#define N_CLS_PAD 48
#define EMBED_DIM 128
#define BATCHN    262144

// workspace layout (floats)
#define OFF_SUMS 0                    // 40*128
#define OFF_CNT  (OFF_SUMS + N_CLASSES * EMBED_DIM)   // 40
#define OFF_ACC  (OFF_CNT + N_CLASSES)                // 4: radial, z2sum, margin, corr
#define OFF_C    (OFF_ACC + 8)                        // 40*128
#define OFF_C2   (OFF_C + N_CLASSES * EMBED_DIM)      // 40
#define WS_ZERO_COUNT (OFF_ACC + 8)

__device__ __forceinline__ float waveReduceSum(float v) {
    #pragma unroll
    for (int off = 16; off > 0; off >>= 1) v += __shfl_xor(v, off, 32);
    return v;
}

__global__ __launch_bounds__(256) void k_zero(float* __restrict__ ws) {
    int i = blockIdx.x * 256 + threadIdx.x;
    if (i < WS_ZERO_COUNT) ws[i] = 0.0f;
}

// Pass 1: per-class sums & counts. One wave per row, float4 per lane, LDS atomics.
__global__ __launch_bounds__(256) void k_segsum(const float* __restrict__ z,
                                                const int* __restrict__ y,
                                                float* __restrict__ ws) {
    __shared__ float s_sum[N_CLASSES * EMBED_DIM];
    __shared__ float s_cnt[N_CLASSES];
    const int tid = threadIdx.x;
    for (int i = tid; i < N_CLASSES * EMBED_DIM; i += 256) s_sum[i] = 0.0f;
    if (tid < N_CLASSES) s_cnt[tid] = 0.0f;
    __syncthreads();

    const int wave = tid >> 5, lane = tid & 31;
    const int rowBase = blockIdx.x * 1024 + wave * 128;   // 8 waves * 128 rows = 1024 rows/block
    #pragma unroll 4
    for (int i = 0; i < 128; ++i) {
        const int row = rowBase + i;
        const int cls = y[row];
        const float4 zv = *(const float4*)(z + (size_t)row * EMBED_DIM + lane * 4);
        float* dst = s_sum + cls * EMBED_DIM + lane * 4;
        atomicAdd(dst + 0, zv.x);
        atomicAdd(dst + 1, zv.y);
        atomicAdd(dst + 2, zv.z);
        atomicAdd(dst + 3, zv.w);
        if (lane == 0) atomicAdd(&s_cnt[cls], 1.0f);
    }
    __syncthreads();
    for (int i = tid; i < N_CLASSES * EMBED_DIM; i += 256)
        unsafeAtomicAdd(ws + OFF_SUMS + i, s_sum[i]);
    if (tid < N_CLASSES) unsafeAtomicAdd(ws + OFF_CNT + tid, s_cnt[tid]);
}

// Finalize centroids, c2, and compact-loss correction: sum_k(cnt_k*|c_k|^2 - 2*c_k.S_k)
__global__ __launch_bounds__(256) void k_centers(const float* __restrict__ centers,
                                                 const unsigned char* __restrict__ initialized,
                                                 float* __restrict__ ws) {
    const int tid = threadIdx.x, wave = tid >> 5, lane = tid & 31;
    float corr = 0.0f;
    for (int k = wave; k < N_CLASSES; k += 8) {
        const float cnt = ws[OFF_CNT + k];
        const int base = k * EMBED_DIM + lane * 4;
        float c2p = 0.0f, dp = 0.0f;
        #pragma unroll
        for (int j = 0; j < 4; ++j) {
            const float S   = ws[OFF_SUMS + base + j];
            const float ctr = centers[base + j];
            const float mean = S / fmaxf(cnt, 1.0f);
            const float ema  = 0.9f * ctr + 0.1f * mean;     // MOMENTUM = 0.1
            const float nv   = initialized[k] ? ema : mean;
            const float cv   = (cnt > 0.0f) ? nv : ctr;
            ws[OFF_C + base + j] = cv;
            c2p += cv * cv;
            dp  += cv * S;
        }
        c2p = waveReduceSum(c2p);
        dp  = waveReduceSum(dp);
        if (lane == 0) {
            ws[OFF_C2 + k] = c2p;
            corr += cnt * c2p - 2.0f * dp;
        }
    }
    if (lane == 0) unsafeAtomicAdd(ws + OFF_ACC + 3, corr);
}

// Main pass: per-wave 16-row tile. WMMA f32 16x16x4 Gram vs 48 padded classes,
// plus radial loss and sum(|z|^2) from the same streamed A fragments.
__global__ __launch_bounds__(256) void k_loss(const float* __restrict__ z,
                                              const int* __restrict__ y,
                                              const float* __restrict__ target_radii,
                                              const float* __restrict__ margins,
                                              float* __restrict__ ws) {
    __shared__ float s_c[N_CLS_PAD * EMBED_DIM];   // 24 KB, classes 40..47 zero
    __shared__ float s_c2[N_CLS_PAD];
    const int tid = threadIdx.x;
    for (int i = tid; i < N_CLS_PAD * EMBED_DIM; i += 256)
        s_c[i] = (i < N_CLASSES * EMBED_DIM) ? ws[OFF_C + i] : 0.0f;
    if (tid < N_CLS_PAD) s_c2[tid] = (tid < N_CLASSES) ? ws[OFF_C2 + tid] : 0.0f;
    __syncthreads();

    const int wave = tid >> 5, lane = tid & 31;
    const int m16 = lane & 15, hi = lane >> 4, kk = hi * 2;
    const int row0 = (blockIdx.x * 8 + wave) * 16;

    const int   yl  = y[row0 + m16];
    const float trl = target_radii[yl];
    const float mgl = margins[yl];

    v8f acc0 = {}, acc1 = {}, acc2 = {};
    float z2p = 0.0f;

    const float* zrow  = z   + (size_t)(row0 + m16) * EMBED_DIM + kk;
    const float* crow0 = s_c + (size_t)(0  + m16) * EMBED_DIM + kk;
    const float* crow1 = s_c + (size_t)(16 + m16) * EMBED_DIM + kk;
    const float* crow2 = s_c + (size_t)(32 + m16) * EMBED_DIM + kk;

    #pragma unroll 8
    for (int k0 = 0; k0 < EMBED_DIM; k0 += 4) {
        const v2f a  = *(const v2f*)(zrow + k0);
        z2p += a.x * a.x + a.y * a.y;
        const v2f b0 = *(const v2f*)(crow0 + k0);
        const v2f b1 = *(const v2f*)(crow1 + k0);
        const v2f b2 = *(const v2f*)(crow2 + k0);
        acc0 = __builtin_amdgcn_wmma_f32_16x16x4_f32(false, a, false, b0, (short)0, acc0, false, false);
        acc1 = __builtin_amdgcn_wmma_f32_16x16x4_f32(false, a, false, b1, (short)0, acc1, false, false);
        acc2 = __builtin_amdgcn_wmma_f32_16x16x4_f32(false, a, false, b2, (short)0, acc2, false, false);
    }

    // full |z|^2 of row (lane&15), in all lanes
    const float z2full = z2p + __shfl_xor(z2p, 16, 32);

    float radial = 0.0f, z2sum = 0.0f;
    if (hi == 0) {
        z2sum = z2full;
        const float r  = sqrtf(z2full);
        const float d  = r - trl;
        const float ad = fabsf(d);
        radial = (ad < 1.0f) ? 0.5f * d * d : (ad - 0.5f);
    }

    const float c2_0 = s_c2[ 0 + m16];
    const float c2_1 = s_c2[16 + m16];
    const float c2_2 = s_c2[32 + m16];
    const float INF  = __builtin_inff();

    float marg = 0.0f;
    #pragma unroll
    for (int r = 0; r < 8; ++r) {
        const int   src = r + 8 * hi;              // D element (r,lane) holds row m = r + 8*hi
        const float z2m = __shfl(z2full, src, 32);
        const int   ym  = __shfl(yl,     src, 32);
        const float mgm = __shfl(mgl,    src, 32);

        float dmin = INF;
        {
            const float d2 = z2m + c2_0 - 2.0f * acc0[r];
            const float ds = sqrtf(fmaxf(d2, 0.0f));
            if (ym != m16) dmin = fminf(dmin, ds);
        }
        {
            const float d2 = z2m + c2_1 - 2.0f * acc1[r];
            const float ds = sqrtf(fmaxf(d2, 0.0f));
            if (ym != 16 + m16) dmin = fminf(dmin, ds);
        }
        {
            const int n = 32 + m16;
            if (n < N_CLASSES && ym != n) {
                const float d2 = z2m + c2_2 - 2.0f * acc2[r];
                dmin = fminf(dmin, sqrtf(fmaxf(d2, 0.0f)));
            }
        }
        // min over 16 lanes of this half (xor masks < 16 stay within the half)
        #pragma unroll
        for (int off = 8; off > 0; off >>= 1)
            dmin = fminf(dmin, __shfl_xor(dmin, off, 32));
        if (m16 == 0) marg += fmaxf(mgm - dmin, 0.0f);
    }

    const float a0 = waveReduceSum(radial);
    const float a1 = waveReduceSum(z2sum);
    const float a2 = waveReduceSum(marg);
    if (lane == 0) {
        unsafeAtomicAdd(ws + OFF_ACC + 0, a0);
        unsafeAtomicAdd(ws + OFF_ACC + 1, a1);
        unsafeAtomicAdd(ws + OFF_ACC + 2, a2);
    }
}

__global__ void k_final(const float* __restrict__ ws, float* __restrict__ out) {
    const float invB    = 1.0f / (float)BATCHN;
    const float radial  = ws[OFF_ACC + 0] * invB;
    const float compact = (ws[OFF_ACC + 1] + ws[OFF_ACC + 3]) * invB;
    const float margin  = ws[OFF_ACC + 2] * invB;
    out[0] = 1.0f * radial + 0.5f * compact + 1.0f * margin;   // LAMBDA_R, LAMBDA_C, LAMBDA_M
}

extern "C" void kernel_launch(void* const* d_in, const int* in_sizes, int n_in,
                              void* d_out, int out_size, void* d_ws, size_t ws_size,
                              hipStream_t stream) {
    const float*         z    = (const float*)d_in[0];
    const int*           y    = (const int*)d_in[1];
    const float*         ctr  = (const float*)d_in[2];
    const unsigned char* init = (const unsigned char*)d_in[3];
    const float*         tr   = (const float*)d_in[4];
    const float*         mg   = (const float*)d_in[5];
    float* ws  = (float*)d_ws;
    float* out = (float*)d_out;

    k_zero<<<(WS_ZERO_COUNT + 255) / 256, 256, 0, stream>>>(ws);
    k_segsum<<<BATCHN / 1024, 256, 0, stream>>>(z, y, ws);
    k_centers<<<1, 256, 0, stream>>>(ctr, init, ws);
    k_loss<<<(BATCHN / 16) / 8, 256, 0, stream>>>(z, y, tr, mg, ws);
    k_final<<<1, 1, 0, stream>>>(ws, out);
}